// MoERouter_90323162235697
// MI455X (gfx1250) — compile-verified
//
#include <hip/hip_runtime.h>
#include <hip/hip_bf16.h>
#include <math.h>

// ---------------------------------------------------------------------------
// MoE router for MI455X (gfx1250).
// logits = x(16384x4096) @ W^T(4096x8): memory bound (256MB of x @ 23.3TB/s
// ~= 11us floor).  fp32 WMMA 16x16x4 keeps exact reference precision; x is
// streamed once via async global->LDS b128 copies, TRIPLE buffered on
// ASYNCcnt (8KB in flight per wave) to cover HBM latency at full bandwidth.
// Experts 8..15 of the 16-wide WMMA tile are clamped duplicates (n&7) and
// simply never read by the epilogue -> no masking, no divergence.
// ---------------------------------------------------------------------------

typedef __attribute__((ext_vector_type(2))) float v2f;
typedef __attribute__((ext_vector_type(4))) float v4f;
typedef __attribute__((ext_vector_type(8))) float v8f;
typedef __attribute__((ext_vector_type(4))) int   i32x4;

// Pointer types matching __builtin_amdgcn_global_load_async_to_lds_b128
// (vV4i*1 V4i*3 Ii Ii).
typedef __attribute__((address_space(1))) i32x4* gbl_i32x4_p;
typedef __attribute__((address_space(3))) i32x4* lds_i32x4_p;

#define NEXP 8
#define TOPK 2
#define DIM 4096
#define ROWS_PER_WAVE 16
#define NWAVES 4
#define ROWS_PER_WG (ROWS_PER_WAVE * NWAVES)   // 64
#define KC 64                                   // K-chunk in floats (256B/row)
#define LDS_ROW 68                              // +4 pad: 16B aligned, spreads banks
#define CHUNKS (DIM / KC)                       // 64
#define NBUF 3                                  // triple buffer (prefetch depth 2)
#define WAVE_TILE (ROWS_PER_WAVE * LDS_ROW)     // 1088 floats

// ---- async global -> LDS 16B copy (CDNA5 GLOBAL_LOAD_ASYNC_TO_LDS_B128) ----
__device__ __forceinline__ void async_cp16(const float* g, float* l) {
#if __has_builtin(__builtin_amdgcn_global_load_async_to_lds_b128)
  __builtin_amdgcn_global_load_async_to_lds_b128(
      (gbl_i32x4_p)g, (lds_i32x4_p)l, 0, 0);
#else
  *(v4f*)l = *(const v4f*)g;                    // sync fallback
#endif
}

template <int N>
__device__ __forceinline__ void wait_async() {
#if __has_builtin(__builtin_amdgcn_s_wait_asynccnt)
  __builtin_amdgcn_s_wait_asynccnt(N);
#else
  asm volatile("s_wait_asynccnt %0" :: "i"(N) : "memory");
#endif
  asm volatile("" ::: "memory");                // keep LDS reads after the wait
}

// ---- f32 WMMA: D(16x16) = A(16x4) * B(4x16) + C ----
__device__ __forceinline__ v8f wmma4(v2f a, v2f b, v8f c) {
#if __has_builtin(__builtin_amdgcn_wmma_f32_16x16x4_f32)
  return __builtin_amdgcn_wmma_f32_16x16x4_f32(
      /*neg_a=*/false, a, /*neg_b=*/false, b,
      /*c_mod=*/(short)0, c, /*reuse_a=*/false, /*reuse_b=*/false);
#else
  c[0] = fmaf(a[0], b[0], c[0]);                // placeholder (builtin missing)
  c[1] = fmaf(a[1], b[1], c[1]);
  return c;
#endif
}

__global__ __launch_bounds__(NWAVES * 32)
void moe_router_gemm(const float* __restrict__ x, const float* __restrict__ W,
                     float* __restrict__ out_w, float* __restrict__ out_i,
                     float* __restrict__ partial)
{
  // 3 x 4 x 1088 floats = 51 KB.  Epilogue scratch aliases stage buffer 0
  // (safe: separated from all staging traffic by __syncthreads()).
  __shared__ __align__(16) float smem[NBUF * NWAVES * WAVE_TILE];
  float* const ep  = smem;                        // [NWAVES][16*17] logits tiles
  float* const red = smem + NWAVES * (16 * 17);   // [64][16] P|f partials

  const int t    = threadIdx.x;
  const int wave = t >> 5;
  const int lane = t & 31;
  const int n    = lane & 15;   // N (expert) for B/C frags, M row for A frag
  const int hi   = lane >> 4;   // selects K pair (A/B) / M half (C)

  const long rowBase = (long)blockIdx.x * ROWS_PER_WG + (long)wave * ROWS_PER_WAVE;
  const float* xw = x + rowBase * DIM;

  // Stage one K-chunk: 16 rows x 256B.  Lane l, step i -> row 2i+hi, col n:
  // both global (i*2*DIM) and LDS (i*2*LDS_ROW) strides are compile-time.
  auto stage_chunk = [&](int buf, int k0) {
    const float* gs = xw + (long)hi * DIM + k0 + n * 4;
    float* ls = smem + (buf * NWAVES + wave) * WAVE_TILE + hi * LDS_ROW + n * 4;
#pragma unroll
    for (int i = 0; i < 8; ++i)
      async_cp16(gs + (long)i * 2 * DIM, ls + i * 2 * LDS_ROW);
  };

  stage_chunk(0, 0);
  stage_chunk(1, KC);

  v8f acc0 = {}, acc1 = {};
  // B[k][n] = W[n][k]; experts 8..15 clamp to 0..7 (their logits are unread).
  const float* wrow = W + (long)(n & 7) * DIM + hi * 2;

  for (int ch = 0; ch < CHUNKS; ++ch) {
    const int k0 = ch * KC;
    if      (ch + 2 < CHUNKS) { stage_chunk((ch + 2) % NBUF, k0 + 2 * KC); wait_async<16>(); }
    else if (ch + 1 < CHUNKS) { wait_async<8>(); }
    else                      { wait_async<0>(); }

    // Prefetch the 16 B fragments for this chunk (L0/WGP$-resident W).
    const float* bcol = wrow + k0;
    v2f bf[KC / 4];
#pragma unroll
    for (int kk = 0; kk < KC / 4; ++kk)
      bf[kk] = *(const v2f*)(bcol + kk * 4);

    // A frag (16x4 f32 layout): lane m = lane&15, VGPR v holds K = 2*hi + v
    const float* arow = smem + ((ch % NBUF) * NWAVES + wave) * WAVE_TILE
                             + n * LDS_ROW + hi * 2;
#pragma unroll
    for (int kk = 0; kk < KC / 4; ++kk) {
      v2f a = *(const v2f*)(arow + kk * 4);
      if (kk & 1) acc1 = wmma4(a, bf[kk], acc1);
      else        acc0 = wmma4(a, bf[kk], acc0);
    }
  }
  const v8f c = acc0 + acc1;

  __syncthreads();   // all staging reads done before ep aliases stage buf 0

  // C/D layout: VGPR r -> M = 8*hi + r, N = lane&15.  Scatter logits to LDS.
#pragma unroll
  for (int r = 0; r < 8; ++r)
    ep[wave * (16 * 17) + (hi * 8 + r) * 17 + n] = c[r];
  __syncthreads();

  // ---- routing: one thread per row (threads 0..63) ----
  if (t < ROWS_PER_WG) {
    const int wv = t >> 4, lr = t & 15;
    float lg[NEXP];
#pragma unroll
    for (int e = 0; e < NEXP; ++e) lg[e] = ep[wv * (16 * 17) + lr * 17 + e];

    int i1 = 0; float v1 = lg[0];                 // top-1 (first index on ties)
#pragma unroll
    for (int e = 1; e < NEXP; ++e) if (lg[e] > v1) { v1 = lg[e]; i1 = e; }
    int i2 = -1; float v2 = -INFINITY;            // top-2
#pragma unroll
    for (int e = 0; e < NEXP; ++e)
      if (e != i1 && lg[e] > v2) { v2 = lg[e]; i2 = e; }

    const float e2  = expf(v2 - v1);              // softmax over [v1, v2]
    const float inv = 1.0f / (1.0f + e2);
    const long row = (long)blockIdx.x * ROWS_PER_WG + t;
    out_w[row * 2 + 0] = inv;
    out_w[row * 2 + 1] = e2 * inv;
    out_i[row * 2 + 0] = (float)i1;
    out_i[row * 2 + 1] = (float)i2;

    float s = 0.0f, g[NEXP];                      // full softmax for aux P
#pragma unroll
    for (int e = 0; e < NEXP; ++e) { g[e] = expf(lg[e] - v1); s += g[e]; }
    const float invs = 1.0f / s;
#pragma unroll
    for (int e = 0; e < NEXP; ++e) {
      red[t * 16 + e]        = g[e] * invs;             // P partial (gates)
      red[t * 16 + NEXP + e] = (e == i1) ? 1.0f : 0.0f; // f partial (top-1)
    }
  }
  __syncthreads();

  // Deterministic fixed-order tree reduction over the 64 rows
#pragma unroll
  for (int sft = 32; sft > 0; sft >>= 1) {
    if (t < sft) {
#pragma unroll
      for (int j = 0; j < 16; ++j) red[t * 16 + j] += red[(t + sft) * 16 + j];
    }
    __syncthreads();
  }
  if (t < 16) partial[(long)blockIdx.x * 16 + t] = red[t];
}

__global__ void moe_router_finalize(const float* __restrict__ partial,
                                    float* __restrict__ out_aux,
                                    int nwg, float inv_n)
{
  __shared__ float tot[16];
  const int t = threadIdx.x;
  if (t < 16) {
    float s = 0.0f;
    for (int i = 0; i < nwg; ++i) s += partial[(long)i * 16 + t]; // fixed order
    tot[t] = s;
  }
  __syncthreads();
  if (t == 0) {
    float acc = 0.0f;
    for (int e = 0; e < NEXP; ++e)
      acc += (tot[NEXP + e] * inv_n) * (tot[e] * inv_n);  // f_e * P_e
    out_aux[0] = 0.01f * (float)NEXP * acc;
  }
}

extern "C" void kernel_launch(void* const* d_in, const int* in_sizes, int n_in,
                              void* d_out, int out_size, void* d_ws, size_t ws_size,
                              hipStream_t stream)
{
  const float* x = (const float*)d_in[0];
  const float* W = (const float*)d_in[1];
  const int N   = in_sizes[0] / DIM;        // 16384
  const int nwg = N / ROWS_PER_WG;          // 256

  float* out   = (float*)d_out;
  float* out_w = out;                       // expert_weights: N*2
  float* out_i = out + (long)N * TOPK;      // expert_indices (as float): N*2
  float* aux   = out + (long)N * TOPK * 2;  // aux_loss: 1
  float* partial = (float*)d_ws;            // nwg*16 floats = 16 KB scratch

  moe_router_gemm<<<nwg, NWAVES * 32, 0, stream>>>(x, W, out_w, out_i, partial);
  moe_router_finalize<<<1, 32, 0, stream>>>(partial, aux, nwg, 1.0f / (float)N);
}